// GCN_40793599377791
// MI455X (gfx1250) — compile-verified
//
#include <hip/hip_runtime.h>
#include <hip/hip_bf16.h>
#include <stdint.h>

// ---------------- problem constants (match reference) ----------------
#define NN 50000
#define EE 800000
#define FF 256
#define HH 128
#define GG 512

typedef __attribute__((ext_vector_type(2))) float v2f;
typedef __attribute__((ext_vector_type(8))) float v8f;

// ---------------- tiny utility kernels ----------------
__global__ void fill_kernel(float* __restrict__ p, float v, int n) {
    int i = blockIdx.x * blockDim.x + threadIdx.x;
    if (i < n) p[i] = v;
}

__global__ void deg_count_kernel(const long long* __restrict__ dst,
                                 float* __restrict__ deg, int e) {
    int i = blockIdx.x * blockDim.x + threadIdx.x;
    if (i < e) atomicAdd(&deg[(int)dst[i]], 1.0f);
}

__global__ void rsqrt_kernel(const float* __restrict__ deg,
                             float* __restrict__ dinv, int n) {
    int i = blockIdx.x * blockDim.x + threadIdx.x;
    if (i < n) {
        float d = deg[i];
        dinv[i] = (d > 0.0f) ? rsqrtf(d) : 0.0f;
    }
}

// ---------------- WMMA fp32 GEMM (fully templated) ----------------
// out = op(A[M,K] @ B) with B = [K,Nh] (row major) or B^T of [Nh,K] (TRANSB).
// One wave -> one 16x16 output tile. blockDim = 32*(Nh/16). grid = M/16.
// K-loop fully unrolled: K/4 chained V_WMMA_F32_16X16X4_F32 per wave.
template <int K, int Nh, bool TRANSB, bool SCALE, bool BIAS, bool RELU, bool DUAL>
__global__ __launch_bounds__(32 * (Nh / 16))
void wmma_gemm_kernel(const float* __restrict__ A,
                      const float* __restrict__ B,
                      const float* __restrict__ scale_row,
                      const float* __restrict__ bias,
                      float* __restrict__ out,
                      float* __restrict__ out2) {
    constexpr int LDA = K + 4;                 // pad: kills 16-way LDS bank conflicts
    constexpr int T   = 32 * (Nh / 16);
    __shared__ float As[16 * LDA];

    const unsigned row0 = blockIdx.x * 16u;

    // cooperative float4 load of the 16xK A tile (coalesced, ds_store_b128)
    const float4* A4 = (const float4*)(A + row0 * (unsigned)K);
    for (int idx = threadIdx.x; idx < 16 * K / 4; idx += T) {
        int r  = idx / (K / 4);
        int c4 = idx - r * (K / 4);
        *(float4*)&As[r * LDA + c4 * 4] = A4[idx];
    }
    __syncthreads();

    const int lane  = threadIdx.x & 31;
    const int wave  = threadIdx.x >> 5;
    const int m     = lane & 15;               // A row handled by this lane
    const int ncol  = wave * 16 + (lane & 15); // output column for this lane
    const int khalf = (lane >> 4) << 1;        // 0 (lanes 0-15) / 2 (lanes 16-31)

    const float* Ap = &As[m * LDA + khalf];
    v8f c = {};

    if (!TRANSB) {
        const float* Bp = B + (unsigned)khalf * Nh + ncol;
#pragma unroll
        for (int k = 0; k < K; k += 4) {
            v2f a = *(const v2f*)(Ap + k);
            v2f b;
            b.x = Bp[(unsigned)k * Nh];
            b.y = Bp[(unsigned)k * Nh + Nh];
            c = __builtin_amdgcn_wmma_f32_16x16x4_f32(false, a, false, b,
                                                      (short)0, c, false, false);
        }
    } else {
        const float* Bp = B + (unsigned)ncol * K + khalf;  // kk,kk+1 contiguous
#pragma unroll
        for (int k = 0; k < K; k += 4) {
            v2f a = *(const v2f*)(Ap + k);
            v2f b = *(const v2f*)(Bp + k);     // global_load_b64
            c = __builtin_amdgcn_wmma_f32_16x16x4_f32(false, a, false, b,
                                                      (short)0, c, false, false);
        }
    }

    const int rhalf = (lane >> 4) << 3;        // 0 or 8
#pragma unroll
    for (int j = 0; j < 8; ++j) {
        unsigned grow = row0 + (unsigned)(j + rhalf);
        float v = c[j];
        if (BIAS)  v += bias[ncol];
        if (SCALE) v *= scale_row[grow];
        if (RELU)  v  = fmaxf(v, 0.0f);
        unsigned o = grow * (unsigned)Nh + (unsigned)ncol;
        out[o] = v;
        if (DUAL) out2[o] = v;                 // self-loop init of the accumulator
    }
}

// ---------------- edge aggregation: acc[d] += P[s] ----------------
// wave handles one edge (indices broadcast); each lane: float4 gather + 4 atomics
__global__ void edge_agg_kernel(const long long* __restrict__ src,
                                const long long* __restrict__ dst,
                                const float* __restrict__ P,
                                float* __restrict__ acc, int e) {
    int tid = blockIdx.x * blockDim.x + threadIdx.x;
    if (tid >= e * (HH / 4)) return;
    int eid = tid >> 5;                        // 32 float4-chunks per edge
    int ch  = (tid & 31) << 2;
    int s = (int)src[eid];
    int d = (int)dst[eid];
    float4 v = *(const float4*)&P[(unsigned)s * HH + ch];
    float* a = &acc[(unsigned)d * HH + ch];
    atomicAdd(a + 0, v.x);
    atomicAdd(a + 1, v.y);
    atomicAdd(a + 2, v.z);
    atomicAdd(a + 3, v.w);
}

// h[i,c] = relu(acc[i,c] * dinv[i] + b[c])
__global__ void conv_finalize_kernel(const float* __restrict__ acc,
                                     const float* __restrict__ dinv,
                                     const float* __restrict__ bias,
                                     float* __restrict__ h, int n) {
    int tid = blockIdx.x * blockDim.x + threadIdx.x;
    if (tid >= n * HH) return;
    int i = tid >> 7;
    int ccol = tid & (HH - 1);
    h[tid] = fmaxf(acc[tid] * dinv[i] + bias[ccol], 0.0f);
}

// layer-3 fused: v = relu(acc*dinv + b); poolsum[batch[i], c] += v
__global__ void conv_finalize_pool_kernel(const float* __restrict__ acc,
                                          const float* __restrict__ dinv,
                                          const float* __restrict__ bias,
                                          const long long* __restrict__ batch,
                                          float* __restrict__ poolsum, int n) {
    int tid = blockIdx.x * blockDim.x + threadIdx.x;
    if (tid >= n * HH) return;
    int i = tid >> 7;
    int ccol = tid & (HH - 1);
    float v = fmaxf(acc[tid] * dinv[i] + bias[ccol], 0.0f);
    int g = (int)batch[i];
    atomicAdd(&poolsum[(unsigned)g * HH + ccol], v);
}

__global__ void pool_count_kernel(const long long* __restrict__ batch,
                                  float* __restrict__ cnt, int n) {
    int i = blockIdx.x * blockDim.x + threadIdx.x;
    if (i < n) atomicAdd(&cnt[(int)batch[i]], 1.0f);
}

// xbn[g,c] = (poolsum/max(cnt,1)) * (gamma[c] * rsqrt(1+eps)) + beta[c]
__global__ void bn_kernel(const float* __restrict__ poolsum,
                          const float* __restrict__ cnt,
                          const float* __restrict__ gamma,
                          const float* __restrict__ beta,
                          float* __restrict__ xbn) {
    int tid = blockIdx.x * blockDim.x + threadIdx.x;
    if (tid >= GG * HH) return;
    int g = tid >> 7;
    int ccol = tid & (HH - 1);
    float inv_n = 1.0f / fmaxf(cnt[g], 1.0f);
    float bnscale = rsqrtf(1.0f + 1e-5f);
    xbn[tid] = poolsum[tid] * inv_n * (gamma[ccol] * bnscale) + beta[ccol];
}

// out2[g,j] = relu(sum_k o[g,k]*l2_w[j,k] + l2_b[j]),  j in {0,1}
__global__ void l2_kernel(const float* __restrict__ o,
                          const float* __restrict__ w,
                          const float* __restrict__ b,
                          float* __restrict__ out2) {
    int tid = blockIdx.x * blockDim.x + threadIdx.x;
    if (tid >= GG * 2) return;
    int g = tid >> 1, j = tid & 1;
    float s = b[j];
#pragma unroll 8
    for (int k = 0; k < HH / 2; ++k)
        s += o[(unsigned)g * (HH / 2) + k] * w[(unsigned)j * (HH / 2) + k];
    out2[tid] = fmaxf(s, 0.0f);
}

// ---------------- host-side orchestration ----------------
static inline int cdiv(long long a, int b) { return (int)((a + b - 1) / b); }

extern "C" void kernel_launch(void* const* d_in, const int* in_sizes, int n_in,
                              void* d_out, int out_size, void* d_ws, size_t ws_size,
                              hipStream_t stream) {
    (void)in_sizes; (void)n_in; (void)out_size; (void)ws_size;
    const float*     x     = (const float*)d_in[0];
    const long long* eidx  = (const long long*)d_in[1];   // [2,E] int64
    const long long* batch = (const long long*)d_in[3];
    const float* W1 = (const float*)d_in[4];  const float* b1 = (const float*)d_in[5];
    const float* W2 = (const float*)d_in[6];  const float* b2 = (const float*)d_in[7];
    const float* W3 = (const float*)d_in[8];  const float* b3 = (const float*)d_in[9];
    const float* bn_g = (const float*)d_in[10]; const float* bn_b = (const float*)d_in[11];
    const float* l1w = (const float*)d_in[12]; const float* l1b = (const float*)d_in[13];
    const float* l2w = (const float*)d_in[14]; const float* l2b = (const float*)d_in[15];

    const long long* src = eidx;
    const long long* dst = eidx + EE;

    float* ws = (float*)d_ws;
    float* deg     = ws;                                        // N
    float* dinv    = ws + NN;                                   // N
    float* P       = ws + 100000;                               // N*H (16B aligned)
    float* acc     = ws + 100000 + (size_t)NN * HH;             // N*H
    float* hbuf    = ws + 100000 + 2 * (size_t)NN * HH;         // N*H
    float* poolsum = ws + 100000 + 3 * (size_t)NN * HH;         // G*H
    float* cnt     = poolsum + (size_t)GG * HH;                 // G
    float* obuf    = cnt + GG;                                  // G*(H/2)

    float* xbn  = (float*)d_out;                                // [G,H]
    float* out2 = (float*)d_out + (size_t)GG * HH;              // [G,2]

    const int TB = 256;
    const int nEdgeT = cdiv((long long)EE * (HH / 4), TB);
    const int nNH    = cdiv((long long)NN * HH, TB);

    // degree (self loop => init 1.0) and dinv
    fill_kernel<<<cdiv(NN, TB), TB, 0, stream>>>(deg, 1.0f, NN);
    deg_count_kernel<<<cdiv(EE, TB), TB, 0, stream>>>(dst, deg, EE);
    rsqrt_kernel<<<cdiv(NN, TB), TB, 0, stream>>>(deg, dinv, NN);

    // zero pooling accumulators (used by fused layer-3 finalize)
    fill_kernel<<<cdiv(GG * HH, TB), TB, 0, stream>>>(poolsum, 0.0f, GG * HH);
    fill_kernel<<<cdiv(GG, TB), TB, 0, stream>>>(cnt, 0.0f, GG);

    // ---- layer 1: P = (x @ W1) * dinv ; acc = P (dual store) ----
    wmma_gemm_kernel<FF, HH, false, true, false, false, true>
        <<<NN / 16, 32 * (HH / 16), 0, stream>>>(x, W1, dinv, nullptr, P, acc);
    edge_agg_kernel<<<nEdgeT, TB, 0, stream>>>(src, dst, P, acc, EE);
    conv_finalize_kernel<<<nNH, TB, 0, stream>>>(acc, dinv, b1, hbuf, NN);

    // ---- layer 2 ----
    wmma_gemm_kernel<HH, HH, false, true, false, false, true>
        <<<NN / 16, 32 * (HH / 16), 0, stream>>>(hbuf, W2, dinv, nullptr, P, acc);
    edge_agg_kernel<<<nEdgeT, TB, 0, stream>>>(src, dst, P, acc, EE);
    conv_finalize_kernel<<<nNH, TB, 0, stream>>>(acc, dinv, b2, hbuf, NN);

    // ---- layer 3 (finalize fused with mean-pool accumulation) ----
    wmma_gemm_kernel<HH, HH, false, true, false, false, true>
        <<<NN / 16, 32 * (HH / 16), 0, stream>>>(hbuf, W3, dinv, nullptr, P, acc);
    edge_agg_kernel<<<nEdgeT, TB, 0, stream>>>(src, dst, P, acc, EE);
    conv_finalize_pool_kernel<<<nNH, TB, 0, stream>>>(acc, dinv, b3, batch, poolsum, NN);
    pool_count_kernel<<<cdiv(NN, TB), TB, 0, stream>>>(batch, cnt, NN);

    // BN (eval) -> xbn (output 1)
    bn_kernel<<<cdiv(GG * HH, TB), TB, 0, stream>>>(poolsum, cnt, bn_g, bn_b, xbn);

    // o = relu(xbn @ l1_w^T + l1_b)  [512,128]@[128,64] via WMMA (transB path)
    wmma_gemm_kernel<HH, HH / 2, true, false, true, true, false>
        <<<GG / 16, 32 * ((HH / 2) / 16), 0, stream>>>(xbn, l1w, nullptr, l1b, obuf, nullptr);

    // out2 = relu(o @ l2_w^T + l2_b)  (output 2)
    l2_kernel<<<cdiv(GG * 2, TB), TB, 0, stream>>>(obuf, l2w, l2b, out2);
}